// SR_GNN_att_agg_with_onehot_27384711479642
// MI455X (gfx1250) — compile-verified
//
#include <hip/hip_runtime.h>
#include <math.h>

// ---------------------------------------------------------------------------
// SR-GNN forward for MI455X (gfx1250, wave32, WMMA).
// Bandwidth-bound problem (~71 GFLOP vs ~1.1 GB traffic @ 23.3 TB/s).
// All GEMMs run on V_WMMA_F32_16X16X4_F32 (exact fp32). Linear layers
// init+msg are algebraically composed into one 100x771 weight.
// ---------------------------------------------------------------------------

typedef float v2f __attribute__((ext_vector_type(2)));
typedef float v8f __attribute__((ext_vector_type(8)));

#define DEV __device__ __forceinline__

constexpr int N_NODES = 81920;
constexpr int N_SESS  = 4096;
constexpr int NPS     = 20;
constexpr int N_EDGES = 163840;
constexpr int IN_DIM  = 771;   // 20+150+100+500+1
constexpr int KF      = 772;   // padded feature dim (pad col = 0)
constexpr int HID     = 100;
constexpr int HP      = 112;   // padded hidden (7 x 16)
constexpr int N_ITEMS = 50000; // 3125 x 16

DEV v8f vzero() { v8f z = {0.f,0.f,0.f,0.f,0.f,0.f,0.f,0.f}; return z; }

// D = A(16x4) x B(4x16) + C, fp32 WMMA
DEV v8f wmma4(v2f a, v2f b, v8f c) {
  return __builtin_amdgcn_wmma_f32_16x16x4_f32(
      /*neg_a=*/false, a, /*neg_b=*/false, b,
      /*c_mod=*/(short)0, c, /*reuse_a=*/false, /*reuse_b=*/false);
}

// A fragment: 16x4 (MxK). lanes 0-15: M=lane, K=k0,k0+1 ; lanes 16-31: K=k0+2,k0+3
DEV v2f frag_a(const float* A, int lda, int k0, int lane) {
  const float* p = A + (lane & 15) * lda + k0 + ((lane >> 4) << 1);
  v2f r; r.x = p[0]; r.y = p[1]; return r;
}
// B fragment: 4x16 (KxN) from row-major BT[K][N]. lanes 0-15: N=lane, K=k0/k0+1
DEV v2f frag_b(const float* B, int ldb, int k0, int lane) {
  const float* p = B + (k0 + ((lane >> 4) << 1)) * ldb + (lane & 15);
  v2f r; r.x = p[0]; r.y = p[ldb]; return r;
}

DEV float sigm(float x) { return 1.f / (1.f + expf(-x)); }

// on-the-fly concat of the 5 feature tensors, with k==771 -> 0 pad
DEV float feat_at(const float* __restrict__ cat, const float* __restrict__ sub,
                  const float* __restrict__ ele, const float* __restrict__ brd,
                  const float* __restrict__ prc, int node, int k) {
  if (k < 20)  return cat[node * 20 + k];
  if (k < 170) return sub[node * 150 + (k - 20)];
  if (k < 270) return ele[node * 100 + (k - 170)];
  if (k < 770) return brd[node * 500 + (k - 270)];
  if (k == 770) return prc[node];
  return 0.f;
}

// ------------------------------ prep kernels -------------------------------

__global__ void k_zero(float* p, int n) {
  int i = blockIdx.x * blockDim.x + threadIdx.x;
  if (i < n) p[i] = 0.f;
}

// WcT[k][n] = (W_msg @ W_init)[n][k], padded to [772][112], pads = 0
__global__ void k_prep_combT(const float* __restrict__ W_msg,
                             const float* __restrict__ W_init,
                             float* __restrict__ WcT) {
  int idx = blockIdx.x * 256 + threadIdx.x;
  if (idx >= KF * HP) return;
  int k = idx / HP, n = idx % HP;
  float v = 0.f;
  if (k < IN_DIM && n < HID) {
    for (int j = 0; j < 128; ++j) v += W_msg[n * 128 + j] * W_init[j * IN_DIM + k];
  }
  WcT[idx] = v;
}

// bcomb[n] = W_msg[n,:] . b_init + b_msg[n], padded to 112
__global__ void k_prep_bcomb(const float* __restrict__ W_msg,
                             const float* __restrict__ b_init,
                             const float* __restrict__ b_msg,
                             float* __restrict__ bcomb) {
  int n = threadIdx.x;
  if (n >= HP) return;
  float v = 0.f;
  if (n < HID) {
    for (int j = 0; j < 128; ++j) v += W_msg[n * 128 + j] * b_init[j];
    v += b_msg[n];
  }
  bcomb[n] = v;
}

// dst[k][n] (112x112) = src[rowoff+n][k], zero-padded
__global__ void k_tpad(float* __restrict__ dst, const float* __restrict__ src,
                       int rowoff) {
  int idx = blockIdx.x * 256 + threadIdx.x;
  if (idx >= HP * HP) return;
  int k = idx / HP, n = idx % HP;
  dst[idx] = (k < HID && n < HID) ? src[(rowoff + n) * HID + k] : 0.f;
}

// BT_fc[k][n] (112x50000) = W_fc[n][k], k>=100 -> 0
__global__ void k_prep_fcT(const float* __restrict__ W_fc, float* __restrict__ BT) {
  int idx = blockIdx.x * 256 + threadIdx.x;
  if (idx >= HP * N_ITEMS) return;
  int k = idx / N_ITEMS, n = idx % N_ITEMS;
  BT[idx] = (k < HID) ? W_fc[n * HID + k] : 0.f;
}

// ------------------- node GEMM: h0 = feat @ WcT (+bcomb) -------------------
// block: 256 thr (8 waves), 32 nodes. LDS-staged feat chunks of K=112.
// 14 wave-tiles (2 M x 7 N); wave w owns tiles {w, w+8}.

__global__ __launch_bounds__(256) void k_node_gemm(
    const float* __restrict__ cat, const float* __restrict__ sub,
    const float* __restrict__ ele, const float* __restrict__ brd,
    const float* __restrict__ prc, const float* __restrict__ WcT,
    const float* __restrict__ bcomb, float* __restrict__ h0) {
  __shared__ float sa[32 * 113];  // ld 113: conflict-free
  int nb = blockIdx.x * 32;
  int tid = threadIdx.x, lane = tid & 31, w = tid >> 5;
  int t0 = w, t1 = w + 8;
  bool has1 = (t1 < 14);
  int mt0 = t0 / 7, nt0 = t0 % 7, mt1 = t1 / 7, nt1 = t1 % 7;
  v8f acc0 = vzero(), acc1 = vzero();

  for (int kc = 0; kc < KF; kc += 112) {
    int klen = KF - kc; if (klen > 112) klen = 112;  // 112 or 100
    __syncthreads();
    for (int i = tid; i < 32 * klen; i += 256) {
      int r = i / klen, cc = i % klen;
      sa[r * 113 + cc] = feat_at(cat, sub, ele, brd, prc, nb + r, kc + cc);
    }
    __syncthreads();
    int ks = klen >> 2;
    const float* a0 = sa + mt0 * 16 * 113;
    const float* b0 = WcT + nt0 * 16;
    for (int s = 0; s < ks; ++s)
      acc0 = wmma4(frag_a(a0, 113, s * 4, lane),
                   frag_b(b0, HP, kc + s * 4, lane), acc0);
    if (has1) {
      const float* a1 = sa + mt1 * 16 * 113;
      const float* b1 = WcT + nt1 * 16;
      for (int s = 0; s < ks; ++s)
        acc1 = wmma4(frag_a(a1, 113, s * 4, lane),
                     frag_b(b1, HP, kc + s * 4, lane), acc1);
    }
  }
  {
    int col = nt0 * 16 + (lane & 15);
    float bias = bcomb[col];
    int rb = nb + mt0 * 16 + ((lane >> 4) << 3);
    #pragma unroll
    for (int r = 0; r < 8; ++r) h0[(rb + r) * HP + col] = acc0[r] + bias;
  }
  if (has1) {
    int col = nt1 * 16 + (lane & 15);
    float bias = bcomb[col];
    int rb = nb + mt1 * 16 + ((lane >> 4) << 3);
    #pragma unroll
    for (int r = 0; r < 8; ++r) h0[(rb + r) * HP + col] = acc1[r] + bias;
  }
}

// ----------------------------- edge aggregation ----------------------------

__global__ void k_edge(const int* __restrict__ ei, const float* __restrict__ h0,
                       float* __restrict__ msum, float* __restrict__ cnt) {
  int e = blockIdx.x;
  int src = ei[e];
  int dst = ei[N_EDGES + e];
  int c = threadIdx.x;
  if (c < HP) atomicAdd(&msum[dst * HP + c], h0[src * HP + c]);
  if (c == 0) atomicAdd(&cnt[dst], 1.f);
}

__global__ void k_invcnt(float* __restrict__ cnt) {
  int i = blockIdx.x * blockDim.x + threadIdx.x;
  if (i < N_NODES) cnt[i] = 1.f / fmaxf(cnt[i], 1.f);
}

// ---------------- last[s] = feat[20s+19] @ W_last.T + b_last ---------------

__global__ __launch_bounds__(128) void k_last(
    const float* __restrict__ cat, const float* __restrict__ sub,
    const float* __restrict__ ele, const float* __restrict__ brd,
    const float* __restrict__ prc, const float* __restrict__ W_last,
    const float* __restrict__ b_last, float* __restrict__ last) {
  __shared__ float fr[772];
  int s = blockIdx.x, t = threadIdx.x;
  int node = s * NPS + NPS - 1;  // segment_max(arange, repeat) == 20s+19
  for (int i = t; i < 772; i += 128) fr[i] = feat_at(cat, sub, ele, brd, prc, node, i);
  __syncthreads();
  if (t < HP) {
    float v = 0.f;
    if (t < HID) {
      for (int k = 0; k < IN_DIM; ++k) v += fr[k] * W_last[t * IN_DIM + k];
      v += b_last[t];
    }
    last[s * HP + t] = v;
  }
}

// ------------------------------- fused GRU ---------------------------------
// A = [m | h0] (K=224) staged in LDS; R,Z over full K; Ni over m-half,
// Nh over h0-half. All four accumulators share fragment layout, so the
// GRU nonlinearity is pure elementwise on accumulators.

DEV void gru_epi(v8f aR, v8f aZ, v8f aNi, v8f aNh, int nb, int mt, int nt,
                 int lane, const float* __restrict__ b_ih,
                 const float* __restrict__ b_hh, const float* __restrict__ h0,
                 const float* __restrict__ last, float* __restrict__ h1) {
  int col = nt * 16 + (lane & 15);
  float bR = 0.f, bZ = 0.f, bNi = 0.f, bNh = 0.f;
  if (col < HID) {
    bR = b_ih[col] + b_hh[col];
    bZ = b_ih[HID + col] + b_hh[HID + col];
    bNi = b_ih[2 * HID + col];
    bNh = b_hh[2 * HID + col];
  }
  int rb = nb + mt * 16 + ((lane >> 4) << 3);
  #pragma unroll
  for (int r = 0; r < 8; ++r) {
    int row = rb + r;
    float rg = sigm(aR[r] + bR);
    float zg = sigm(aZ[r] + bZ);
    float ng = tanhf(aNi[r] + bNi + rg * (aNh[r] + bNh));
    float hv = (1.f - zg) * ng + zg * h0[row * HP + col] + last[(row / NPS) * HP + col];
    h1[row * HP + col] = (col < HID) ? hv : 0.f;
  }
}

__global__ __launch_bounds__(256) void k_gru(
    const float* __restrict__ msum, const float* __restrict__ invc,
    const float* __restrict__ h0, const float* __restrict__ BT_R,
    const float* __restrict__ BT_Z, const float* __restrict__ BT_Ni,
    const float* __restrict__ BT_Nh, const float* __restrict__ b_ih,
    const float* __restrict__ b_hh, const float* __restrict__ last,
    float* __restrict__ h1) {
  __shared__ float sa[32 * 225];
  int nb = blockIdx.x * 32;
  int tid = threadIdx.x, lane = tid & 31, w = tid >> 5;
  for (int i = tid; i < 32 * 224; i += 256) {
    int r = i / 224, kk = i % 224;
    int node = nb + r;
    sa[r * 225 + kk] = (kk < HP) ? msum[node * HP + kk] * invc[node]
                                 : h0[node * HP + (kk - HP)];
  }
  __syncthreads();
  int t0 = w, t1 = w + 8;
  bool has1 = (t1 < 14);
  int mt0 = t0 / 7, nt0 = t0 % 7, mt1 = t1 / 7, nt1 = t1 % 7;
  v8f aR0 = vzero(), aZ0 = vzero(), aNi0 = vzero(), aNh0 = vzero();
  v8f aR1 = vzero(), aZ1 = vzero(), aNi1 = vzero(), aNh1 = vzero();

  for (int s = 0; s < 56; ++s) {
    int k0 = s * 4;
    {
      v2f a = frag_a(sa + mt0 * 16 * 225, 225, k0, lane);
      aR0 = wmma4(a, frag_b(BT_R + nt0 * 16, HP, k0, lane), aR0);
      aZ0 = wmma4(a, frag_b(BT_Z + nt0 * 16, HP, k0, lane), aZ0);
      if (k0 < HP) aNi0 = wmma4(a, frag_b(BT_Ni + nt0 * 16, HP, k0, lane), aNi0);
      else         aNh0 = wmma4(a, frag_b(BT_Nh + nt0 * 16, HP, k0 - HP, lane), aNh0);
    }
    if (has1) {
      v2f a = frag_a(sa + mt1 * 16 * 225, 225, k0, lane);
      aR1 = wmma4(a, frag_b(BT_R + nt1 * 16, HP, k0, lane), aR1);
      aZ1 = wmma4(a, frag_b(BT_Z + nt1 * 16, HP, k0, lane), aZ1);
      if (k0 < HP) aNi1 = wmma4(a, frag_b(BT_Ni + nt1 * 16, HP, k0, lane), aNi1);
      else         aNh1 = wmma4(a, frag_b(BT_Nh + nt1 * 16, HP, k0 - HP, lane), aNh1);
    }
  }
  gru_epi(aR0, aZ0, aNi0, aNh0, nb, mt0, nt0, lane, b_ih, b_hh, h0, last, h1);
  if (has1) gru_epi(aR1, aZ1, aNi1, aNh1, nb, mt1, nt1, lane, b_ih, b_hh, h0, last, h1);
}

// ------------------- q = relu(h1 @ W_g1.T + b_g1) (WMMA) -------------------

__global__ __launch_bounds__(256) void k_gate(
    const float* __restrict__ h1, const float* __restrict__ BT_g1,
    const float* __restrict__ b_g1, float* __restrict__ q) {
  __shared__ float sa[32 * 113];
  int nb = blockIdx.x * 32;
  int tid = threadIdx.x, lane = tid & 31, w = tid >> 5;
  for (int i = tid; i < 32 * HP; i += 256) {
    int r = i / HP, kk = i % HP;
    sa[r * 113 + kk] = h1[(nb + r) * HP + kk];
  }
  __syncthreads();
  int t0 = w, t1 = w + 8;
  bool has1 = (t1 < 14);
  int mt0 = t0 / 7, nt0 = t0 % 7, mt1 = t1 / 7, nt1 = t1 % 7;
  v8f a0 = vzero(), a1 = vzero();
  for (int s = 0; s < 28; ++s) {
    int k0 = s * 4;
    a0 = wmma4(frag_a(sa + mt0 * 16 * 113, 113, k0, lane),
               frag_b(BT_g1 + nt0 * 16, HP, k0, lane), a0);
    if (has1)
      a1 = wmma4(frag_a(sa + mt1 * 16 * 113, 113, k0, lane),
                 frag_b(BT_g1 + nt1 * 16, HP, k0, lane), a1);
  }
  {
    int col = nt0 * 16 + (lane & 15);
    float bias = (col < HID) ? b_g1[col] : 0.f;
    int rb = nb + mt0 * 16 + ((lane >> 4) << 3);
    #pragma unroll
    for (int r = 0; r < 8; ++r)
      q[(rb + r) * HP + col] = (col < HID) ? fmaxf(a0[r] + bias, 0.f) : 0.f;
  }
  if (has1) {
    int col = nt1 * 16 + (lane & 15);
    float bias = (col < HID) ? b_g1[col] : 0.f;
    int rb = nb + mt1 * 16 + ((lane >> 4) << 3);
    #pragma unroll
    for (int r = 0; r < 8; ++r)
      q[(rb + r) * HP + col] = (col < HID) ? fmaxf(a1[r] + bias, 0.f) : 0.f;
  }
}

// ----------------------- g[node] = q[node] . W_g2 + b ----------------------

__global__ __launch_bounds__(256) void k_g(const float* __restrict__ q,
                                           const float* __restrict__ W_g2,
                                           const float* __restrict__ b_g2,
                                           float* __restrict__ g) {
  int w = threadIdx.x >> 5, lane = threadIdx.x & 31;
  int node = blockIdx.x * 8 + w;
  float v = 0.f;
  for (int k = lane; k < HID; k += 32) v += q[node * HP + k] * W_g2[k];
  v += __shfl_xor(v, 16, 32);
  v += __shfl_xor(v, 8, 32);
  v += __shfl_xor(v, 4, 32);
  v += __shfl_xor(v, 2, 32);
  v += __shfl_xor(v, 1, 32);
  if (lane == 0) g[node] = v + b_g2[0];
}

// ----------------- per-session softmax(20) + weighted pool -----------------

__global__ __launch_bounds__(64) void k_attn(const float* __restrict__ g,
                                             const float* __restrict__ h1,
                                             float* __restrict__ pooled) {
  __shared__ float sg[NPS], sal[NPS], sred[2];
  int s = blockIdx.x, t = threadIdx.x;
  if (t < NPS) sg[t] = g[s * NPS + t];
  __syncthreads();
  if (t == 0) {
    float gm = -3.4e38f;
    for (int i = 0; i < NPS; ++i) gm = fmaxf(gm, sg[i]);
    float den = 0.f;
    for (int i = 0; i < NPS; ++i) den += expf(sg[i] - gm);
    sred[0] = gm; sred[1] = den;
  }
  __syncthreads();
  if (t < NPS) sal[t] = expf(sg[t] - sred[0]) / sred[1];
  __syncthreads();
  for (int c = t; c < HP; c += 64) {
    float v = 0.f;
    for (int i = 0; i < NPS; ++i) v += sal[i] * h1[(s * NPS + i) * HP + c];
    pooled[s * HP + c] = v;  // pad cols auto-zero (h1 pads are 0)
  }
}

// ---------------- final FC: out = pooled @ W_fc.T + b_fc -------------------
// Store-bound (819 MB out); BT_fc (22 MB) and pooled (1.8 MB) are L2-hot.

__global__ __launch_bounds__(256) void k_fc(const float* __restrict__ pooled,
                                            const float* __restrict__ BT_fc,
                                            const float* __restrict__ b_fc,
                                            float* __restrict__ out) {
  int w = threadIdx.x >> 5, lane = threadIdx.x & 31;
  int nt = blockIdx.x * 8 + w;
  if (nt >= N_ITEMS / 16) return;  // wave-uniform guard (EXEC stays all-1s)
  int mb = blockIdx.y * 16;
  const float* ab = pooled + mb * HP;
  const float* bb = BT_fc + nt * 16;
  v8f acc = vzero();
  #pragma unroll
  for (int s = 0; s < 28; ++s)
    acc = wmma4(frag_a(ab, HP, s * 4, lane), frag_b(bb, N_ITEMS, s * 4, lane), acc);
  int col = nt * 16 + (lane & 15);
  float bias = b_fc[col];
  int rb = mb + ((lane >> 4) << 3);
  #pragma unroll
  for (int r = 0; r < 8; ++r)
    out[(size_t)(rb + r) * N_ITEMS + col] = acc[r] + bias;
}

// -------------------------------- launcher ---------------------------------

extern "C" void kernel_launch(void* const* d_in, const int* in_sizes, int n_in,
                              void* d_out, int out_size, void* d_ws, size_t ws_size,
                              hipStream_t stream) {
  const float* cat    = (const float*)d_in[0];
  const float* sub    = (const float*)d_in[1];
  const float* ele    = (const float*)d_in[2];
  const float* brd    = (const float*)d_in[3];
  const float* prc    = (const float*)d_in[4];
  const int*   eidx   = (const int*)d_in[5];
  const float* W_init = (const float*)d_in[7];
  const float* b_init = (const float*)d_in[8];
  const float* W_msg  = (const float*)d_in[9];
  const float* b_msg  = (const float*)d_in[10];
  const float* W_ih   = (const float*)d_in[11];
  const float* W_hh   = (const float*)d_in[12];
  const float* b_ih   = (const float*)d_in[13];
  const float* b_hh   = (const float*)d_in[14];
  const float* W_last = (const float*)d_in[15];
  const float* b_last = (const float*)d_in[16];
  const float* W_g1   = (const float*)d_in[17];
  const float* b_g1   = (const float*)d_in[18];
  const float* W_g2   = (const float*)d_in[19];
  const float* b_g2   = (const float*)d_in[20];
  const float* W_fc   = (const float*)d_in[21];
  const float* b_fc   = (const float*)d_in[22];
  float* out = (float*)d_out;

  float* ws   = (float*)d_ws;
  float* WcT  = ws;               // 772*112
  float* bcmb = WcT  + KF * HP;   // 112
  float* BT_R = bcmb + HP;        // 224*112
  float* BT_Z = BT_R + 2 * HP * HP;
  float* BT_Ni = BT_Z + 2 * HP * HP;  // 112*112
  float* BT_Nh = BT_Ni + HP * HP;
  float* BT_g1 = BT_Nh + HP * HP;
  float* BT_fc = BT_g1 + HP * HP;     // 112*50000
  float* h0   = BT_fc + HP * N_ITEMS; // 81920*112 (reused as q after GRU)
  float* msum = h0   + N_NODES * HP;
  float* h1   = msum + N_NODES * HP;
  float* cnt  = h1   + N_NODES * HP;  // 81920 (becomes invcnt)
  float* gbuf = cnt  + N_NODES;       // 81920
  float* last = gbuf + N_NODES;       // 4096*112
  float* pool = last + N_SESS * HP;   // 4096*112

  const int nmsum = N_NODES * HP;
  k_zero<<<(nmsum + 255) / 256, 256, 0, stream>>>(msum, nmsum);
  k_zero<<<(N_NODES + 255) / 256, 256, 0, stream>>>(cnt, N_NODES);

  k_prep_combT<<<(KF * HP + 255) / 256, 256, 0, stream>>>(W_msg, W_init, WcT);
  k_prep_bcomb<<<1, 128, 0, stream>>>(W_msg, b_init, b_msg, bcmb);
  k_tpad<<<49, 256, 0, stream>>>(BT_R,            W_ih, 0);
  k_tpad<<<49, 256, 0, stream>>>(BT_R + HP * HP,  W_hh, 0);
  k_tpad<<<49, 256, 0, stream>>>(BT_Z,            W_ih, 100);
  k_tpad<<<49, 256, 0, stream>>>(BT_Z + HP * HP,  W_hh, 100);
  k_tpad<<<49, 256, 0, stream>>>(BT_Ni,           W_ih, 200);
  k_tpad<<<49, 256, 0, stream>>>(BT_Nh,           W_hh, 200);
  k_tpad<<<49, 256, 0, stream>>>(BT_g1,           W_g1, 0);
  k_prep_fcT<<<(HP * N_ITEMS + 255) / 256, 256, 0, stream>>>(W_fc, BT_fc);

  k_node_gemm<<<N_NODES / 32, 256, 0, stream>>>(cat, sub, ele, brd, prc, WcT, bcmb, h0);
  k_edge<<<N_EDGES, 128, 0, stream>>>(eidx, h0, msum, cnt);
  k_invcnt<<<(N_NODES + 255) / 256, 256, 0, stream>>>(cnt);
  k_last<<<N_SESS, 128, 0, stream>>>(cat, sub, ele, brd, prc, W_last, b_last, last);
  k_gru<<<N_NODES / 32, 256, 0, stream>>>(msum, cnt, h0, BT_R, BT_Z, BT_Ni, BT_Nh,
                                          b_ih, b_hh, last, h1);
  k_gate<<<N_NODES / 32, 256, 0, stream>>>(h1, BT_g1, b_g1, h0 /*q*/);
  k_g<<<N_NODES / 8, 256, 0, stream>>>(h0, W_g2, b_g2, gbuf);
  k_attn<<<N_SESS, 64, 0, stream>>>(gbuf, h1, pool);
  k_fc<<<dim3((N_ITEMS / 16 + 7) / 8, N_SESS / 16), 256, 0, stream>>>(pool, BT_fc, b_fc, out);
}